// PressureProjection_37855841747352
// MI455X (gfx1250) — compile-verified
//
#include <hip/hip_runtime.h>
#include <math.h>

// ---------------------------------------------------------------------------
// Pressure projection: divergence -> Laplacian -> 20-iter Jacobi-PCG -> correct.
// Memory-bound sparse workload. Per CG iter: stream (src,dst,w)=96MB; since
// src+dst+w+node vectors ~110MB < 192MB L2, iterations are largely L2-resident.
// WMMA f32 16x16x4 with B=ones serves as a full-precision wave32 reduction
// engine for the dot products. Graph nodes minimized (5/iteration) since
// replay overhead competes with the ~85us HBM floor.
// ---------------------------------------------------------------------------

#define BLK 256
#define CG_ITERS 20
#define CG_TOL 1e-4f

typedef float v2f __attribute__((ext_vector_type(2)));
typedef float v8f __attribute__((ext_vector_type(8)));

// Wave32 sum via V_WMMA_F32_16X16X4_F32: A holds one partial per lane (a1=0),
// B = ones => D[m][n] = sum_k A[m][k]. Sum of the 8 D VGPRs + xor-16 shuffle
// yields the grand total in every lane. EXEC must be all 1s here.
__device__ __forceinline__ float wave_sum32(float v) {
#if __has_builtin(__builtin_amdgcn_wmma_f32_16x16x4_f32)
  v2f a; a[0] = v;    a[1] = 0.0f;
  v2f b; b[0] = 1.0f; b[1] = 1.0f;
  v8f c = {};
  c = __builtin_amdgcn_wmma_f32_16x16x4_f32(false, a, false, b, (short)0, c,
                                            false, false);
  float s = c[0] + c[1] + c[2] + c[3] + c[4] + c[5] + c[6] + c[7];
  s += __shfl_xor(s, 16);
  return s;
#else
  for (int off = 16; off > 0; off >>= 1) v += __shfl_xor(v, off);
  return v;
#endif
}

// Block-wide sum, atomically added to *dst_scalar by thread 0.
// Must be reached by ALL threads of the block (wave-uniform control flow).
__device__ __forceinline__ void block_atomic_sum(float v, float* dst_scalar) {
  __shared__ float lds[BLK / 32];
  float s = wave_sum32(v);
  int lane = threadIdx.x & 31;
  int wid  = threadIdx.x >> 5;
  if (lane == 0) lds[wid] = s;
  __syncthreads();
  if (threadIdx.x == 0) {
    float t = 0.f;
    for (int i = 0; i < (int)(blockDim.x >> 5); ++i) t += lds[i];
    atomicAdd(dst_scalar, t);
  }
  __syncthreads();  // safe reuse of lds by a second call
}

// Scalar slots: 0=rz 1=rr 2=pAp 3=rz_new 4=rr_new 5=alpha(unused) 6=beta
// 7=active 8=done
#define S_RZ 0
#define S_RR 1
#define S_PAP 2
#define S_RZN 3
#define S_RRN 4
#define S_BETA 6
#define S_ACT 7
#define S_DONE 8

// ---------------------------- setup kernels --------------------------------

__global__ void k_init(float* diag, float* r, float* phi, float* sc, int n) {
  int i = blockIdx.x * blockDim.x + threadIdx.x;
  if (i < n) { diag[i] = 0.f; r[i] = 0.f; phi[i] = 0.f; }
  if (i == 0) { for (int j = 0; j < 16; ++j) sc[j] = 0.f; }
}

__device__ __forceinline__ void setup_one(int s, int d, float a, float h,
                                          float nx, float ny, float nz,
                                          const float* __restrict__ uh,
                                          float* wslot, float* diag,
                                          float* fluxsum) {
  float we = a / fmaxf(h, 1e-8f);
  if (wslot) *wslot = we;
  float fx = 0.5f * (uh[3 * s + 0] + uh[3 * d + 0]);
  float fy = 0.5f * (uh[3 * s + 1] + uh[3 * d + 1]);
  float fz = 0.5f * (uh[3 * s + 2] + uh[3 * d + 2]);
  atomicAdd(&fluxsum[d], (fx * nx + fy * ny + fz * nz) * a);
  atomicAdd(&diag[d], we);
}

__global__ void k_edge_setup(const float* __restrict__ uh,
                             const int* __restrict__ src,
                             const int* __restrict__ dst,
                             const float* __restrict__ nrm,
                             const float* __restrict__ area,
                             const float* __restrict__ dist,
                             float* __restrict__ w, float* diag, float* fluxsum,
                             int e, int vec) {
  int t = blockIdx.x * blockDim.x + threadIdx.x;
  int base = 4 * t;
  if (vec && base + 3 < e) {
    int4   s4 = *(const int4*)(src + base);
    int4   d4 = *(const int4*)(dst + base);
    float4 a4 = *(const float4*)(area + base);
    float4 h4 = *(const float4*)(dist + base);
    float nv[12];
    *(float4*)(nv + 0) = *(const float4*)(nrm + 3 * base + 0);
    *(float4*)(nv + 4) = *(const float4*)(nrm + 3 * base + 4);
    *(float4*)(nv + 8) = *(const float4*)(nrm + 3 * base + 8);
    int   ss[4] = {s4.x, s4.y, s4.z, s4.w};
    int   dd[4] = {d4.x, d4.y, d4.z, d4.w};
    float aa[4] = {a4.x, a4.y, a4.z, a4.w};
    float hh[4] = {h4.x, h4.y, h4.z, h4.w};
#pragma unroll
    for (int k = 0; k < 4; ++k)
      setup_one(ss[k], dd[k], aa[k], hh[k], nv[3 * k], nv[3 * k + 1],
                nv[3 * k + 2], uh, w ? w + base + k : nullptr, diag, fluxsum);
  } else {
    for (int i = base; i < e && i < base + 4; ++i)
      setup_one(src[i], dst[i], area[i], dist[i], nrm[3 * i], nrm[3 * i + 1],
                nrm[3 * i + 2], uh, w ? w + i : nullptr, diag, fluxsum);
  }
}

// Per-node: b = flux/vol; r=b; m_inv; z=m_inv*b; p=z; reduce rz0, rr0.
__global__ void k_node_init(const float* __restrict__ vol,
                            const float* __restrict__ diag, float* minv,
                            float* r, float* z, float* p, float* sc, int n,
                            int vec) {
  int t = blockIdx.x * blockDim.x + threadIdx.x;
  int base = 4 * t;
  float rz_p = 0.f, rr_p = 0.f;
  if (vec && base + 3 < n) {
    float fl[4], vv[4], dg[4], bb[4], mi[4], zz[4];
    *(float4*)fl = *(const float4*)(r + base);
    *(float4*)vv = *(const float4*)(vol + base);
    *(float4*)dg = *(const float4*)(diag + base);
#pragma unroll
    for (int k = 0; k < 4; ++k) {
      bb[k] = fl[k] / fmaxf(vv[k], 1e-12f);
      mi[k] = 1.f / fmaxf(dg[k], 1e-8f);
      zz[k] = mi[k] * bb[k];
      rz_p += bb[k] * zz[k];
      rr_p += bb[k] * bb[k];
    }
    *(float4*)(r + base)    = *(float4*)bb;
    *(float4*)(minv + base) = *(float4*)mi;
    *(float4*)(z + base)    = *(float4*)zz;
    *(float4*)(p + base)    = *(float4*)zz;
  } else {
    for (int i = base; i < n && i < base + 4; ++i) {
      float b  = r[i] / fmaxf(vol[i], 1e-12f);
      float mi = 1.f / fmaxf(diag[i], 1e-8f);
      float zv = mi * b;
      r[i] = b; minv[i] = mi; z[i] = zv; p[i] = zv;
      rz_p += b * zv;
      rr_p += b * b;
    }
  }
  block_atomic_sum(rz_p, &sc[S_RZ]);
  block_atomic_sum(rr_p, &sc[S_RR]);
}

__global__ void k_begin0(float* sc) {  // <<<1,1>>>
  sc[S_ACT] = (sc[S_DONE] == 0.f && sc[S_RR] >= CG_TOL * CG_TOL) ? 1.f : 0.f;
  sc[S_PAP] = 0.f; sc[S_RZN] = 0.f; sc[S_RRN] = 0.f;
}

__global__ void k_ap_diag(const float* __restrict__ diag,
                          const float* __restrict__ p, float* ap,
                          const float* __restrict__ sc, int n, int vec) {
  if (sc[S_ACT] == 0.f) return;
  int t = blockIdx.x * blockDim.x + threadIdx.x;
  int base = 4 * t;
  if (vec && base + 3 < n) {
    float dg[4], pp[4], av[4];
    *(float4*)dg = *(const float4*)(diag + base);
    *(float4*)pp = *(const float4*)(p + base);
#pragma unroll
    for (int k = 0; k < 4; ++k) av[k] = dg[k] * pp[k];
    *(float4*)(ap + base) = *(float4*)av;
  } else {
    for (int i = base; i < n && i < base + 4; ++i) ap[i] = diag[i] * p[i];
  }
}

// ---------------------------- CG iteration ---------------------------------

__global__ void k_ap_edge(const int* __restrict__ src,
                          const int* __restrict__ dst,
                          const float* __restrict__ w,
                          const float* __restrict__ area,
                          const float* __restrict__ dist,
                          const float* __restrict__ p, float* __restrict__ ap,
                          const float* __restrict__ sc, int e, int vec) {
  if (sc[S_ACT] == 0.f) return;
  int t = blockIdx.x * blockDim.x + threadIdx.x;
  int base = 4 * t;
  if (vec && base + 3 < e) {
    int4 s4 = *(const int4*)(src + base);
    int4 d4 = *(const int4*)(dst + base);
    float ww[4];
    if (w) {
      *(float4*)ww = *(const float4*)(w + base);
    } else {
      float aa[4], hh[4];
      *(float4*)aa = *(const float4*)(area + base);
      *(float4*)hh = *(const float4*)(dist + base);
#pragma unroll
      for (int k = 0; k < 4; ++k) ww[k] = aa[k] / fmaxf(hh[k], 1e-8f);
    }
    int ss[4] = {s4.x, s4.y, s4.z, s4.w};
    int dd[4] = {d4.x, d4.y, d4.z, d4.w};
#pragma unroll
    for (int k = 0; k < 4; ++k) atomicAdd(&ap[dd[k]], -ww[k] * p[ss[k]]);
  } else {
    for (int i = base; i < e && i < base + 4; ++i) {
      float we = w ? w[i] : area[i] / fmaxf(dist[i], 1e-8f);
      atomicAdd(&ap[dst[i]], -we * p[src[i]]);
    }
  }
}

__global__ void k_dot_pap(const float* __restrict__ p,
                          const float* __restrict__ ap, float* sc, int n,
                          int vec) {
  if (sc[S_ACT] == 0.f) return;
  int t = blockIdx.x * blockDim.x + threadIdx.x;
  int base = 4 * t;
  float v = 0.f;
  if (vec && base + 3 < n) {
    float pp[4], av[4];
    *(float4*)pp = *(const float4*)(p + base);
    *(float4*)av = *(const float4*)(ap + base);
#pragma unroll
    for (int k = 0; k < 4; ++k) v += pp[k] * av[k];
  } else {
    for (int i = base; i < n && i < base + 4; ++i) v += p[i] * ap[i];
  }
  block_atomic_sum(v, &sc[S_PAP]);
}

// alpha computed per-thread from scalar slots; updates phi,r,z; reduces rz,rr.
__global__ void k_update_xr(const float* __restrict__ minv,
                            const float* __restrict__ ap,
                            const float* __restrict__ pv, float* phi, float* r,
                            float* z, float* sc, int n, int vec) {
  if (sc[S_ACT] == 0.f) return;  // uniform: whole grid exits together
  float alpha = sc[S_RZ] / (sc[S_PAP] + 1e-12f);
  int t = blockIdx.x * blockDim.x + threadIdx.x;
  int base = 4 * t;
  float rz_p = 0.f, rr_p = 0.f;
  if (vec && base + 3 < n) {
    float ph[4], pp[4], rr[4], av[4], mi[4], zz[4];
    *(float4*)ph = *(const float4*)(phi + base);
    *(float4*)pp = *(const float4*)(pv + base);
    *(float4*)rr = *(const float4*)(r + base);
    *(float4*)av = *(const float4*)(ap + base);
    *(float4*)mi = *(const float4*)(minv + base);
#pragma unroll
    for (int k = 0; k < 4; ++k) {
      ph[k] += alpha * pp[k];
      rr[k] -= alpha * av[k];
      zz[k] = mi[k] * rr[k];
      rz_p += rr[k] * zz[k];
      rr_p += rr[k] * rr[k];
    }
    *(float4*)(phi + base) = *(float4*)ph;
    *(float4*)(r + base)   = *(float4*)rr;
    *(float4*)(z + base)   = *(float4*)zz;
  } else {
    for (int i = base; i < n && i < base + 4; ++i) {
      phi[i] += alpha * pv[i];
      float rv = r[i] - alpha * ap[i];
      r[i] = rv;
      float zv = minv[i] * rv;
      z[i] = zv;
      rz_p += rv * zv;
      rr_p += rv * rv;
    }
  }
  block_atomic_sum(rz_p, &sc[S_RZN]);
  block_atomic_sum(rr_p, &sc[S_RRN]);
}

// Fused end-of-iteration + begin-of-next scalar logic.  <<<1,1>>>
__global__ void k_scal(float* sc) {
  if (sc[S_ACT] != 0.f) {
    sc[S_BETA] = sc[S_RZN] / (sc[S_RZ] + 1e-12f);
    sc[S_RZ]   = sc[S_RZN];
    sc[S_RR]   = sc[S_RRN];
    sc[S_DONE] = 0.f;
  } else {
    sc[S_DONE] = 1.f;
  }
  sc[S_ACT] = (sc[S_DONE] == 0.f && sc[S_RR] >= CG_TOL * CG_TOL) ? 1.f : 0.f;
  sc[S_PAP] = 0.f; sc[S_RZN] = 0.f; sc[S_RRN] = 0.f;
}

// p = z + beta*p fused with next iteration's ap = diag*p.
// Guarded by NEXT iteration's act: if inactive, p/ap are never consumed again.
__global__ void k_update_p_ap(const float* __restrict__ z,
                              const float* __restrict__ diag, float* p,
                              float* ap, const float* __restrict__ sc, int n,
                              int vec) {
  if (sc[S_ACT] == 0.f) return;
  float beta = sc[S_BETA];
  int t = blockIdx.x * blockDim.x + threadIdx.x;
  int base = 4 * t;
  if (vec && base + 3 < n) {
    float zz[4], pp[4], dg[4], av[4];
    *(float4*)zz = *(const float4*)(z + base);
    *(float4*)pp = *(const float4*)(p + base);
    *(float4*)dg = *(const float4*)(diag + base);
#pragma unroll
    for (int k = 0; k < 4; ++k) {
      pp[k] = zz[k] + beta * pp[k];
      av[k] = dg[k] * pp[k];
    }
    *(float4*)(p + base)  = *(float4*)pp;
    *(float4*)(ap + base) = *(float4*)av;
  } else {
    for (int i = base; i < n && i < base + 4; ++i) {
      float pn = z[i] + beta * p[i];
      p[i]  = pn;
      ap[i] = diag[i] * pn;
    }
  }
}

// ---------------------------- correction -----------------------------------

__global__ void k_copy_u(const float* __restrict__ uh, float* out_u, int n3,
                         int vec) {
  int t = blockIdx.x * blockDim.x + threadIdx.x;
  int base = 4 * t;
  if (vec && base + 3 < n3) {
    *(float4*)(out_u + base) = *(const float4*)(uh + base);
  } else {
    for (int i = base; i < n3 && i < base + 4; ++i) out_u[i] = uh[i];
  }
}

__device__ __forceinline__ void grad_one(int s, int d, float we, float nx,
                                         float ny, float nz,
                                         const float* __restrict__ vol,
                                         const float* __restrict__ phi,
                                         float* out_u) {
  float c = we * phi[s] / fmaxf(vol[d], 1e-12f);
  atomicAdd(&out_u[3 * d + 0], -c * nx);
  atomicAdd(&out_u[3 * d + 1], -c * ny);
  atomicAdd(&out_u[3 * d + 2], -c * nz);
}

__global__ void k_grad_edge(const int* __restrict__ src,
                            const int* __restrict__ dst,
                            const float* __restrict__ w,
                            const float* __restrict__ area,
                            const float* __restrict__ dist,
                            const float* __restrict__ nrm,
                            const float* __restrict__ vol,
                            const float* __restrict__ phi, float* out_u, int e,
                            int vec) {
  int t = blockIdx.x * blockDim.x + threadIdx.x;
  int base = 4 * t;
  if (vec && base + 3 < e) {
    int4 s4 = *(const int4*)(src + base);
    int4 d4 = *(const int4*)(dst + base);
    float ww[4];
    if (w) {
      *(float4*)ww = *(const float4*)(w + base);
    } else {
      float aa[4], hh[4];
      *(float4*)aa = *(const float4*)(area + base);
      *(float4*)hh = *(const float4*)(dist + base);
#pragma unroll
      for (int k = 0; k < 4; ++k) ww[k] = aa[k] / fmaxf(hh[k], 1e-8f);
    }
    float nv[12];
    *(float4*)(nv + 0) = *(const float4*)(nrm + 3 * base + 0);
    *(float4*)(nv + 4) = *(const float4*)(nrm + 3 * base + 4);
    *(float4*)(nv + 8) = *(const float4*)(nrm + 3 * base + 8);
    int ss[4] = {s4.x, s4.y, s4.z, s4.w};
    int dd[4] = {d4.x, d4.y, d4.z, d4.w};
#pragma unroll
    for (int k = 0; k < 4; ++k)
      grad_one(ss[k], dd[k], ww[k], nv[3 * k], nv[3 * k + 1], nv[3 * k + 2],
               vol, phi, out_u);
  } else {
    for (int i = base; i < e && i < base + 4; ++i) {
      float we = w ? w[i] : area[i] / fmaxf(dist[i], 1e-8f);
      grad_one(src[i], dst[i], we, nrm[3 * i], nrm[3 * i + 1], nrm[3 * i + 2],
               vol, phi, out_u);
    }
  }
}

// ---------------------------- host launch ----------------------------------

extern "C" void kernel_launch(void* const* d_in, const int* in_sizes, int n_in,
                              void* d_out, int out_size, void* d_ws,
                              size_t ws_size, hipStream_t stream) {
  const float* u_hat = (const float*)d_in[0];
  const int*   eidx  = (const int*)d_in[1];
  const float* nrm   = (const float*)d_in[2];
  const float* area  = (const float*)d_in[3];
  const float* dist  = (const float*)d_in[4];
  const float* vol   = (const float*)d_in[5];

  const int E = in_sizes[3];  // face_areas: (E,)
  const int N = in_sizes[5];  // cell_volumes: (N,)

  float* out_u = (float*)d_out;                   // (N,3) flat
  float* phi   = (float*)d_out + 3 * (size_t)N;   // (N,)

  // Workspace: [0..63] scalars/pad, 6 node vectors, optional edge weights.
  float* f    = (float*)d_ws;
  float* sc   = f;
  float* diag = f + 64;
  float* minv = diag + N;
  float* r    = minv + N;
  float* z    = r + N;
  float* p    = z + N;
  float* ap   = p + N;
  size_t need_w = (64 + 6 * (size_t)N + (size_t)E) * sizeof(float);
  float* w = (ws_size >= need_w) ? (ap + N) : nullptr;

  const int* src = eidx;
  const int* dst = eidx + E;

  // Vector-path legality: all b128 base pointers 16B aligned.
  uintptr_t me = (uintptr_t)src | (uintptr_t)dst | (uintptr_t)area |
                 (uintptr_t)dist | (uintptr_t)nrm | (uintptr_t)u_hat |
                 (w ? (uintptr_t)w : 0);
  uintptr_t mn = (uintptr_t)diag | (uintptr_t)minv | (uintptr_t)r |
                 (uintptr_t)z | (uintptr_t)p | (uintptr_t)ap |
                 (uintptr_t)phi | (uintptr_t)vol | (uintptr_t)out_u;
  int vecE = ((me & 15) == 0) ? 1 : 0;
  int vecN = ((mn & 15) == 0) ? 1 : 0;

  dim3 blk(BLK);
  dim3 gN(((N + 3) / 4 + BLK - 1) / BLK);
  dim3 gN1((N + BLK - 1) / BLK);
  dim3 gN3(((3 * N + 3) / 4 + BLK - 1) / BLK);
  dim3 gE(((E + 3) / 4 + BLK - 1) / BLK);

  // Setup
  k_init<<<gN1, blk, 0, stream>>>(diag, r, phi, sc, N);
  k_edge_setup<<<gE, blk, 0, stream>>>(u_hat, src, dst, nrm, area, dist, w,
                                       diag, r, E, vecE);
  k_node_init<<<gN, blk, 0, stream>>>(vol, diag, minv, r, z, p, sc, N, vecN);
  k_begin0<<<1, 1, 0, stream>>>(sc);
  k_ap_diag<<<gN, blk, 0, stream>>>(diag, p, ap, sc, N, vecN);

  // 20 CG iterations, 5 graph nodes each (3 on the last).
  for (int it = 0; it < CG_ITERS; ++it) {
    k_ap_edge<<<gE, blk, 0, stream>>>(src, dst, w, area, dist, p, ap, sc, E,
                                      vecE);
    k_dot_pap<<<gN, blk, 0, stream>>>(p, ap, sc, N, vecN);
    k_update_xr<<<gN, blk, 0, stream>>>(minv, ap, p, phi, r, z, sc, N, vecN);
    if (it + 1 < CG_ITERS) {
      k_scal<<<1, 1, 0, stream>>>(sc);
      k_update_p_ap<<<gN, blk, 0, stream>>>(z, diag, p, ap, sc, N, vecN);
    }
  }

  // Correction: u_corrected = u_hat - grad_phi
  k_copy_u<<<gN3, blk, 0, stream>>>(u_hat, out_u, 3 * N, vecN);
  k_grad_edge<<<gE, blk, 0, stream>>>(src, dst, w, area, dist, nrm, vol, phi,
                                      out_u, E, vecE);
}